// GraphConvPatternRecognitionModel_30932354466237
// MI455X (gfx1250) — compile-verified
//
#include <hip/hip_runtime.h>
#include <hip/hip_bf16.h>

typedef __attribute__((ext_vector_type(2))) float v2f;
typedef __attribute__((ext_vector_type(8))) float v8f;

#define N_NODES_C 100000
#define HID_C 80

// ---------------------------------------------------------------------------
// Stage 1: edge scatter for layer 1 (6-dim payload: aggregate BEFORE transform)
// agg1[dst] += ew * x[src]   (x is [N,6])
// ---------------------------------------------------------------------------
__global__ void edge_agg1_kernel(const float* __restrict__ x,
                                 const long long* __restrict__ ei,
                                 const float* __restrict__ ew,
                                 float* __restrict__ agg1, int E) {
    int e = blockIdx.x * blockDim.x + threadIdx.x;
    if (e >= E) return;
    long long src = ei[e];
    long long dst = ei[(long long)E + e];
    float w = ew[e];
    const float* xs = x + src * 6;
    float* a = agg1 + dst * 6;
#pragma unroll
    for (int f = 0; f < 6; ++f) atomicAdd(&a[f], w * xs[f]);
}

// ---------------------------------------------------------------------------
// Stage 2: h1 = relu([agg1 | x] @ [W1_rel ; W1_root] + b1)  via WMMA f32 16x16x4
// One wave computes a 16(M) x 80(N) tile. K = 12 (3 steps of 4).
// ---------------------------------------------------------------------------
__device__ __forceinline__ float getA1(const float* agg1, const float* x,
                                       int node, int feat) {
    return (feat < 6) ? agg1[node * 6 + feat] : x[node * 6 + (feat - 6)];
}
__device__ __forceinline__ float getW1(const float* Wrel, const float* Wroot,
                                       int k, int n) {
    return (k < 6) ? Wrel[k * HID_C + n] : Wroot[(k - 6) * HID_C + n];
}

__global__ void layer1_wmma_kernel(const float* __restrict__ x,
                                   const float* __restrict__ agg1,
                                   const float* __restrict__ W_rel,
                                   const float* __restrict__ W_root,
                                   const float* __restrict__ bias,
                                   float* __restrict__ h1, int n_nodes) {
    int wave = (blockIdx.x * blockDim.x + threadIdx.x) >> 5;   // global wave id
    int lane = threadIdx.x & 31;
    int n_tiles = (n_nodes + 15) >> 4;
    if (wave >= n_tiles) return;          // wave-uniform: EXEC stays all-ones
    int m0 = wave << 4;
    int half = lane >> 4;                 // 0: K pair {0,1}; 1: K pair {2,3}
    int l = lane & 15;
    int arow = m0 + l;

    v8f c[5];
#pragma unroll
    for (int t = 0; t < 5; ++t) c[t] = (v8f)(0.0f);

#pragma unroll
    for (int s = 0; s < 3; ++s) {
        int k0 = s * 4 + half * 2;
        v2f A = { getA1(agg1, x, arow, k0), getA1(agg1, x, arow, k0 + 1) };
#pragma unroll
        for (int t = 0; t < 5; ++t) {
            int n = t * 16 + l;
            v2f B = { getW1(W_rel, W_root, k0, n), getW1(W_rel, W_root, k0 + 1, n) };
            c[t] = __builtin_amdgcn_wmma_f32_16x16x4_f32(
                false, A, false, B, (short)0, c[t], false, false);
        }
    }

    // D layout: VGPR r, lanes 0-15 -> (M=r, N=l); lanes 16-31 -> (M=8+r, N=l)
#pragma unroll
    for (int t = 0; t < 5; ++t) {
        int col = t * 16 + l;
        float bv = bias[col];
#pragma unroll
        for (int r = 0; r < 8; ++r) {
            int row = m0 + r + half * 8;
            float v = c[t][r] + bv;
            h1[row * HID_C + col] = v > 0.0f ? v : 0.0f;
        }
    }
}

// ---------------------------------------------------------------------------
// Stage 3: edge scatter for layer 2 (80-dim). 16 lanes per edge, 5 feats/lane
// -> 64B-coalesced gathers and L2 atomics (agg2 is 32MB, L2-resident).
// ---------------------------------------------------------------------------
__global__ void edge_agg2_kernel(const float* __restrict__ h1,
                                 const long long* __restrict__ ei,
                                 const float* __restrict__ ew,
                                 float* __restrict__ agg2, int E) {
    int gid = blockIdx.x * blockDim.x + threadIdx.x;
    int e = gid >> 4;
    int l = gid & 15;
    if (e >= E) return;
    long long src = ei[e];
    long long dst = ei[(long long)E + e];
    float w = ew[e];
    const float* hs = h1 + src * HID_C;
    float* a = agg2 + dst * HID_C;
#pragma unroll
    for (int i = 0; i < 5; ++i) {
        int f = l + 16 * i;
        atomicAdd(&a[f], w * hs[f]);
    }
}

// ---------------------------------------------------------------------------
// Stage 4: h2 = relu(agg2 @ W2_rel + h1 @ W2_root + b2) via WMMA, K = 160.
// Combined weights (160x80 = 50KB) staged in LDS once per block.
// h2 is written IN PLACE over agg2 (each wave touches only its own 16 rows,
// and all its loads complete into registers before its stores).
// ---------------------------------------------------------------------------
__global__ void layer2_wmma_kernel(const float* __restrict__ h1,
                                   float* __restrict__ agg2_h2,
                                   const float* __restrict__ W_rel,
                                   const float* __restrict__ W_root,
                                   const float* __restrict__ bias,
                                   int n_nodes) {
    __shared__ float wlds[160 * HID_C];   // 51,200 bytes
    for (int i = threadIdx.x; i < 160 * HID_C; i += blockDim.x)
        wlds[i] = (i < 80 * HID_C) ? W_rel[i] : W_root[i - 80 * HID_C];
    __syncthreads();

    int wave = (blockIdx.x * blockDim.x + threadIdx.x) >> 5;
    int lane = threadIdx.x & 31;
    int n_tiles = (n_nodes + 15) >> 4;
    if (wave >= n_tiles) return;          // wave-uniform
    int m0 = wave << 4;
    int half = lane >> 4;
    int l = lane & 15;
    int arow = m0 + l;

    const float* aggRow = agg2_h2 + arow * HID_C;
    const float* h1Row  = h1 + arow * HID_C;

    v8f c[5];
#pragma unroll
    for (int t = 0; t < 5; ++t) c[t] = (v8f)(0.0f);

#pragma unroll 4
    for (int s = 0; s < 40; ++s) {        // K = 160, 4 per step
        int k0 = s * 4 + half * 2;        // even; pair never crosses the 80-split
        const float* ab = (k0 < 80) ? (aggRow + k0) : (h1Row + (k0 - 80));
        v2f A = { ab[0], ab[1] };
#pragma unroll
        for (int t = 0; t < 5; ++t) {
            int n = t * 16 + l;
            v2f B = { wlds[k0 * HID_C + n], wlds[(k0 + 1) * HID_C + n] };
            c[t] = __builtin_amdgcn_wmma_f32_16x16x4_f32(
                false, A, false, B, (short)0, c[t], false, false);
        }
    }

#pragma unroll
    for (int t = 0; t < 5; ++t) {
        int col = t * 16 + l;
        float bv = bias[col];
#pragma unroll
        for (int r = 0; r < 8; ++r) {
            int row = m0 + r + half * 8;
            float v = c[t][r] + bv;
            agg2_h2[row * HID_C + col] = v > 0.0f ? v : 0.0f;  // h2 in place
        }
    }
}

// ---------------------------------------------------------------------------
// Stage 5: out[g] = b_out + sum_{i in g} h2[i] @ W_out   (pool fused with GEMM)
// ---------------------------------------------------------------------------
__global__ void out_init_kernel(const float* __restrict__ b_out,
                                float* __restrict__ out, int total) {
    int i = blockIdx.x * blockDim.x + threadIdx.x;
    if (i < total) out[i] = b_out[i % 7];
}

__global__ void pool_out_kernel(const float* __restrict__ h2,
                                const long long* __restrict__ batch,
                                const float* __restrict__ W_out,
                                float* __restrict__ out, int n_nodes) {
    int i = blockIdx.x * blockDim.x + threadIdx.x;
    if (i >= n_nodes) return;
    long long g = batch[i];
    const float* h = h2 + (long long)i * HID_C;
    float s[7] = {0.f, 0.f, 0.f, 0.f, 0.f, 0.f, 0.f};
#pragma unroll 8
    for (int f = 0; f < HID_C; ++f) {
        float v = h[f];
#pragma unroll
        for (int o = 0; o < 7; ++o) s[o] += v * W_out[f * 7 + o];
    }
#pragma unroll
    for (int o = 0; o < 7; ++o) atomicAdd(&out[g * 7 + o], s[o]);
}

// ---------------------------------------------------------------------------
extern "C" void kernel_launch(void* const* d_in, const int* in_sizes, int n_in,
                              void* d_out, int out_size, void* d_ws, size_t ws_size,
                              hipStream_t stream) {
    const float*     x       = (const float*)d_in[0];
    const long long* ei      = (const long long*)d_in[1];   // [2, E] int64
    const float*     ew      = (const float*)d_in[2];
    const long long* batch   = (const long long*)d_in[3];
    const float*     W1_rel  = (const float*)d_in[4];
    const float*     W1_root = (const float*)d_in[5];
    const float*     b1      = (const float*)d_in[6];
    const float*     W2_rel  = (const float*)d_in[7];
    const float*     W2_root = (const float*)d_in[8];
    const float*     b2      = (const float*)d_in[9];
    const float*     W_out   = (const float*)d_in[10];
    const float*     b_out   = (const float*)d_in[11];
    float* out = (float*)d_out;

    const int N = in_sizes[0] / 6;       // 100000
    const int E = in_sizes[2];           // 1000000

    // Workspace layout (256B aligned): agg1 [N,6] | h1 [N,80] | agg2/h2 [N,80]
    char* ws = (char*)d_ws;
    size_t off = 0;
    float* agg1 = (float*)(ws + off); off += ((size_t)N * 6 * 4 + 255) & ~(size_t)255;
    float* h1   = (float*)(ws + off); off += ((size_t)N * HID_C * 4 + 255) & ~(size_t)255;
    float* agg2 = (float*)(ws + off);  // h2 reuses this buffer in place
    (void)ws_size; (void)n_in;

    hipMemsetAsync(agg1, 0, (size_t)N * 6 * sizeof(float), stream);
    hipMemsetAsync(agg2, 0, (size_t)N * HID_C * sizeof(float), stream);

    // Layer 1 scatter
    edge_agg1_kernel<<<(E + 255) / 256, 256, 0, stream>>>(x, ei, ew, agg1, E);

    // Layer 1 node update (WMMA): one wave per 16-node tile, 4 waves/block
    int tiles = (N + 15) / 16;
    int blocks_l = (tiles + 3) / 4;
    layer1_wmma_kernel<<<blocks_l, 128, 0, stream>>>(x, agg1, W1_rel, W1_root, b1, h1, N);

    // Layer 2 scatter: 16 lanes per edge
    long long t2 = (long long)E * 16;
    edge_agg2_kernel<<<(int)((t2 + 255) / 256), 256, 0, stream>>>(h1, ei, ew, agg2, E);

    // Layer 2 node update (WMMA, weights in LDS), h2 in place over agg2
    layer2_wmma_kernel<<<blocks_l, 128, 0, stream>>>(h1, agg2, W2_rel, W2_root, b2, N);

    // Pool + output GEMM fused
    out_init_kernel<<<(out_size + 255) / 256, 256, 0, stream>>>(b_out, out, out_size);
    pool_out_kernel<<<(N + 255) / 256, 256, 0, stream>>>(agg2, batch, W_out, out, N);
}